// simpleGNN_49933289783569
// MI455X (gfx1250) — compile-verified
//
#include <hip/hip_runtime.h>
#include <hip/hip_bf16.h>
#include <math.h>

#define N_NODES    50000
#define N_EDGES    640000
#define DIM        128
#define NUM_GRAPHS 512
#define LDS_STRIDE 132   // 128 + 4 floats pad -> conflict-free b64 LDS reads

typedef float v2f __attribute__((ext_vector_type(2)));
typedef float v8f __attribute__((ext_vector_type(8)));

// ---------------------------------------------------------------------------
// 1) Edge scatter: aggr[dst] += x[src] * w    (one wave per edge, 4 f32/lane)
// ---------------------------------------------------------------------------
__global__ void gnn_scatter(const float* __restrict__ feat,
                            const int*   __restrict__ src,
                            const int*   __restrict__ dst,
                            const float* __restrict__ ew,
                            float*       __restrict__ aggr) {
    const int e = blockIdx.x * (blockDim.x >> 5) + (threadIdx.x >> 5);
    if (e >= N_EDGES) return;
    const int lane = threadIdx.x & 31;
    const int s = src[e];
    const int d = dst[e];
    const float w = ew[e];
    const float4 v = ((const float4*)(feat + (size_t)s * DIM))[lane];
    float* out = aggr + (size_t)d * DIM + lane * 4;
    atomicAdd(out + 0, v.x * w);
    atomicAdd(out + 1, v.y * w);
    atomicAdd(out + 2, v.z * w);
    atomicAdd(out + 3, v.w * w);
}

// ---------------------------------------------------------------------------
// 2) Fused GraphConv GEMM:  Out = relu(Aggr @ Wrel + Root @ Wroot + bias)
//    One block = one 16-row M-tile; 8 waves = 8 N-tiles (full 128 cols).
//    A tiles staged in LDS (coalesced float4 global loads, padded rows),
//    fp32 WMMA 16x16x4, K = 128 (32 steps per operand matrix).
// ---------------------------------------------------------------------------
__global__ void __launch_bounds__(256)
gnn_graphconv_wmma(const float* __restrict__ Aggr,   // [N,128]
                   const float* __restrict__ Root,   // [N,128]
                   const float* __restrict__ Wrel,   // [128,128]
                   const float* __restrict__ Wroot,  // [128,128]
                   const float* __restrict__ bias,   // [128]
                   float*       __restrict__ Out,    // [N,128]
                   int nrows) {
    __shared__ float sA[16 * LDS_STRIDE];            // Aggr tile, padded rows
    __shared__ float sR[16 * LDS_STRIDE];            // Root tile, padded rows

    const int row0 = blockIdx.x * 16;
    if (row0 >= nrows) return;                       // block-uniform; never taken

    const int tid = threadIdx.x;

    // Cooperative stage: 16 rows x 32 float4 = 512 float4, 2 per thread
    #pragma unroll
    for (int i = tid; i < 16 * (DIM / 4); i += 256) {
        const int r  = i >> 5;                       // / (DIM/4)
        const int c4 = i & 31;                       // % (DIM/4)
        const float4 va = ((const float4*)(Aggr + (size_t)(row0 + r) * DIM))[c4];
        const float4 vr = ((const float4*)(Root + (size_t)(row0 + r) * DIM))[c4];
        *(float4*)&sA[r * LDS_STRIDE + c4 * 4] = va;
        *(float4*)&sR[r * LDS_STRIDE + c4 * 4] = vr;
    }
    __syncthreads();

    const int wave = tid >> 5;                       // N-tile index 0..7
    const int lane = tid & 31;
    const int half = lane >> 4;                      // K sub-pair selector
    const int l16  = lane & 15;
    const int bcol = wave * 16 + l16;                // global output column

    const float* sa = &sA[l16 * LDS_STRIDE];         // this lane's A row
    const float* sr = &sR[l16 * LDS_STRIDE];

    v8f c = {0.f, 0.f, 0.f, 0.f, 0.f, 0.f, 0.f, 0.f};

    #pragma unroll 4
    for (int k = 0; k < DIM; k += 4) {
        const int ka = k + 2 * half;
        // Aggr @ Wrel
        const v2f a1 = *(const v2f*)&sa[ka];         // ds_load_b64, conflict-free
        const v2f b1 = { Wrel[(size_t)ka * DIM + bcol],
                         Wrel[(size_t)(ka + 1) * DIM + bcol] };
        c = __builtin_amdgcn_wmma_f32_16x16x4_f32(false, a1, false, b1,
                                                  (short)0, c, false, false);
        // Root @ Wroot
        const v2f a2 = *(const v2f*)&sr[ka];
        const v2f b2 = { Wroot[(size_t)ka * DIM + bcol],
                         Wroot[(size_t)(ka + 1) * DIM + bcol] };
        c = __builtin_amdgcn_wmma_f32_16x16x4_f32(false, a2, false, b2,
                                                  (short)0, c, false, false);
    }

    // C/D layout: lane L, vgpr r -> row = row0 + r + 8*(L/16), col = bcol
    const float bv = bias[bcol];
    #pragma unroll
    for (int r = 0; r < 8; ++r) {
        float v = c[r] + bv;
        v = v > 0.f ? v : 0.f;                       // ReLU
        Out[(size_t)(row0 + r + 8 * half) * DIM + bcol] = v;
    }
}

// ---------------------------------------------------------------------------
// 3) Segment-max pooling. Inputs are post-ReLU (>= 0) so fp32 bits are
//    monotone as uint -> atomicMax on uint with 0-initialized accumulators.
// ---------------------------------------------------------------------------
__global__ void gnn_pool_max(const float* __restrict__ H,
                             const int*   __restrict__ batch,
                             unsigned int* __restrict__ pooled) {  // [G*128]
    const int node = blockIdx.x * (blockDim.x >> 5) + (threadIdx.x >> 5);
    if (node >= N_NODES) return;
    const int lane = threadIdx.x & 31;
    const int g = batch[node];
    const float4 v = ((const float4*)(H + (size_t)node * DIM))[lane];
    unsigned int* p = pooled + (size_t)g * DIM + lane * 4;
    atomicMax(p + 0, __float_as_uint(v.x));
    atomicMax(p + 1, __float_as_uint(v.y));
    atomicMax(p + 2, __float_as_uint(v.z));
    atomicMax(p + 3, __float_as_uint(v.w));
}

// ---------------------------------------------------------------------------
// 4) Final linear (128 -> 1) + sigmoid. One wave32 per graph.
// ---------------------------------------------------------------------------
__global__ void gnn_final(const float* __restrict__ pooled,  // [G,128]
                          const float* __restrict__ Wlin,    // [128]
                          const float* __restrict__ blin,    // [1]
                          float*       __restrict__ out) {   // [G]
    const int g = blockIdx.x * (blockDim.x >> 5) + (threadIdx.x >> 5);
    if (g >= NUM_GRAPHS) return;
    const int lane = threadIdx.x & 31;
    const float4 p = ((const float4*)(pooled + (size_t)g * DIM))[lane];
    const float4 w = ((const float4*)Wlin)[lane];
    float s = p.x * w.x + p.y * w.y + p.z * w.z + p.w * w.w;
    #pragma unroll
    for (int off = 16; off > 0; off >>= 1)
        s += __shfl_down(s, off, 32);
    if (lane == 0) {
        const float z = s + blin[0];
        out[g] = 1.f / (1.f + expf(-z));
    }
}

// ---------------------------------------------------------------------------
// Host launcher
// ---------------------------------------------------------------------------
extern "C" void kernel_launch(void* const* d_in, const int* in_sizes, int n_in,
                              void* d_out, int out_size, void* d_ws, size_t ws_size,
                              hipStream_t stream) {
    const float* x      = (const float*)d_in[0];
    const int*   eidx   = (const int*)  d_in[1];   // [2, E]
    const int*   batch  = (const int*)  d_in[2];
    const float* ew     = (const float*)d_in[3];
    const float* W1rel  = (const float*)d_in[4];
    const float* b1rel  = (const float*)d_in[5];
    const float* W1root = (const float*)d_in[6];
    const float* W2rel  = (const float*)d_in[7];
    const float* b2rel  = (const float*)d_in[8];
    const float* W2root = (const float*)d_in[9];
    const float* Wlin   = (const float*)d_in[10];
    const float* blin   = (const float*)d_in[11];

    const int* src = eidx;
    const int* dst = eidx + N_EDGES;

    // Workspace layout
    const size_t nodeElems = (size_t)N_NODES * DIM;
    float*        aggr   = (float*)d_ws;                       // reused both layers
    float*        h1     = aggr + nodeElems;
    float*        h2     = h1 + nodeElems;
    unsigned int* pooled = (unsigned int*)(h2 + nodeElems);    // [G*128]

    const int scatterBlocks = N_EDGES / 8;                     // 8 edges / 256-thr block
    const int gemmBlocks    = N_NODES / 16;                    // 1 M-tile / block
    const int poolBlocks    = (N_NODES + 7) / 8;               // 8 nodes / 256-thr block
    const int finalBlocks   = NUM_GRAPHS / 8;                  // 8 graphs / 256-thr block

    // ---- Layer 1 ----
    hipMemsetAsync(aggr, 0, nodeElems * sizeof(float), stream);
    gnn_scatter<<<scatterBlocks, 256, 0, stream>>>(x, src, dst, ew, aggr);
    gnn_graphconv_wmma<<<gemmBlocks, 256, 0, stream>>>(aggr, x, W1rel, W1root,
                                                       b1rel, h1, N_NODES);

    // ---- Layer 2 ----
    hipMemsetAsync(aggr, 0, nodeElems * sizeof(float), stream);
    gnn_scatter<<<scatterBlocks, 256, 0, stream>>>(h1, src, dst, ew, aggr);
    gnn_graphconv_wmma<<<gemmBlocks, 256, 0, stream>>>(aggr, h1, W2rel, W2root,
                                                       b2rel, h2, N_NODES);

    // ---- Pool + classify ----
    hipMemsetAsync(pooled, 0, (size_t)NUM_GRAPHS * DIM * sizeof(unsigned int), stream);
    gnn_pool_max<<<poolBlocks, 256, 0, stream>>>(h2, batch, pooled);
    gnn_final<<<finalBlocks, 256, 0, stream>>>((const float*)pooled, Wlin, blin,
                                               (float*)d_out);
}